// GetCostVolume_13932873908710
// MI455X (gfx1250) — compile-verified
//
#include <hip/hip_runtime.h>
#include <math.h>

// ---------------------------------------------------------------------------
// GetCostVolume for MI455X (gfx1250, wave32).
//
// Stage 1: four 1x1 convs (32x32 GEMM per pixel) via V_WMMA_F32_16X16X4_F32,
//          writing Q/K in NHWC layout into d_ws (4 x 33.5 MB = 134 MB,
//          L2-resident on MI455X's 192 MB L2).
// Stage 2: cost volume: per (b,h,w,dir) thread, 24 disparities, 4-corner
//          bilinear gather of contiguous 128B channel vectors + dot/32.
//
// Cache policy: all stream-once traffic (x/y reads, disp reads, Q reads,
// cost-output writes) is non-temporal so the 192 MB L2 is reserved for the
// K tensors, which see heavy random reuse from the bilinear gathers.
// ---------------------------------------------------------------------------

typedef __attribute__((ext_vector_type(2))) float v2f;
typedef __attribute__((ext_vector_type(4))) float v4f;
typedef __attribute__((ext_vector_type(8))) float v8f;

#define B_   2
#define C_   32
#define H_   256
#define W_   512
#define ND_  24
#define HW_  (H_ * W_)            // 131072 = 2^17
#define NPIX_ (B_ * HW_)          // 262144
#define NPC_ ((size_t)NPIX_ * C_) // 8388608 floats per NHWC tensor

// ---------------------------------------------------------------------------
// Stage 1: conv1x1 via fp32 WMMA.
// A (16x4)  = weight tile:  lane<16 -> W[row][4kk+0..1], lane>=16 -> [4kk+2..3]
// B (4x16)  = activations:  elem0 = X[4kk+2*hi][pix], elem1 = X[4kk+2*hi+1][pix]
// C (16x16) = VGPR r: lanes0-15 M=r, lanes16-31 M=r+8; N = lane&15 (pixel col)
// ---------------------------------------------------------------------------
__device__ __forceinline__ void gemm_tile_store(
    const v2f* __restrict__ Bv,         // 8 K-step activation fragments
    const float* __restrict__ Wm,       // [32,32] row-major weight
    const float* __restrict__ bias,     // [32]
    float* __restrict__ outT,           // NHWC output tensor base
    int pix, int col, int hi)
{
#pragma unroll
  for (int mt = 0; mt < 2; ++mt) {
    v8f acc = {0.f, 0.f, 0.f, 0.f, 0.f, 0.f, 0.f, 0.f};
    const int row = mt * 16 + col;
    const float* wr = Wm + row * C_;
#pragma unroll
    for (int kk = 0; kk < 8; ++kk) {
      v2f a;
      a.x = wr[4 * kk + 2 * hi + 0];
      a.y = wr[4 * kk + 2 * hi + 1];
      acc = __builtin_amdgcn_wmma_f32_16x16x4_f32(
          /*neg_a=*/false, a, /*neg_b=*/false, Bv[kk],
          /*c_mod=*/(short)0, acc, /*reuse_a=*/false, /*reuse_b=*/false);
    }
    const int obase = mt * 16 + hi * 8;   // this lane's first output channel
    const float* bp = bias + obase;
    float* op = outT + (size_t)pix * C_ + obase;
    v4f v0, v1;
    v0.x = acc[0] + bp[0]; v0.y = acc[1] + bp[1];
    v0.z = acc[2] + bp[2]; v0.w = acc[3] + bp[3];
    v1.x = acc[4] + bp[4]; v1.y = acc[5] + bp[5];
    v1.z = acc[6] + bp[6]; v1.w = acc[7] + bp[7];
    // Regular-temporal stores: Q/K must stay resident in L2 for stage 2.
    *(v4f*)(op)     = v0;
    *(v4f*)(op + 4) = v1;
  }
}

__global__ void __launch_bounds__(256) conv1x1_wmma_kernel(
    const float* __restrict__ x, const float* __restrict__ y,
    const float* __restrict__ Wq, const float* __restrict__ bq,
    const float* __restrict__ Wk, const float* __restrict__ bk,
    float* __restrict__ ws)
{
  const int lane = threadIdx.x & 31;
  const int wave = threadIdx.x >> 5;
  const int col  = lane & 15;
  const int hi   = lane >> 4;
  const int pix  = ((blockIdx.x * 8 + wave) << 4) + col;  // exact grid, no mask
  const int b    = pix >> 17;            // HW_ = 2^17
  const int hw   = pix & (HW_ - 1);

  const float* xb = x + (size_t)b * C_ * HW_ + hw;
  const float* yb = y + (size_t)b * C_ * HW_ + hw;

  // x/y are each consumed exactly once -> non-temporal loads (TH=NT),
  // keeping L2 free for the NHWC Q/K working set.
  v2f bxv[8], byv[8];
#pragma unroll
  for (int kk = 0; kk < 8; ++kk) {
    const int c0 = 4 * kk + 2 * hi;
    bxv[kk].x = __builtin_nontemporal_load(xb + (size_t)c0 * HW_);
    bxv[kk].y = __builtin_nontemporal_load(xb + (size_t)(c0 + 1) * HW_);
    byv[kk].x = __builtin_nontemporal_load(yb + (size_t)c0 * HW_);
    byv[kk].y = __builtin_nontemporal_load(yb + (size_t)(c0 + 1) * HW_);
  }

  float* Qx = ws;                 // dir0 query  (conv(x, Wq))
  float* Ky = ws + NPC_;          // dir0 key    (conv(y, Wk))
  float* Qy = ws + 2 * NPC_;      // dir1 query  (conv(y, Wq))
  float* Kx = ws + 3 * NPC_;      // dir1 key    (conv(x, Wk))

  gemm_tile_store(bxv, Wq, bq, Qx, pix, col, hi);
  gemm_tile_store(byv, Wk, bk, Ky, pix, col, hi);
  gemm_tile_store(byv, Wq, bq, Qy, pix, col, hi);
  gemm_tile_store(bxv, Wk, bk, Kx, pix, col, hi);
}

// ---------------------------------------------------------------------------
// Stage 2: cost volume with zero-padded bilinear sampling of NHWC K.
// ---------------------------------------------------------------------------
__device__ __forceinline__ float corner_dot(const float* __restrict__ Kb,
                                            const v4f* __restrict__ q,
                                            int yc, int xc, float wgt, bool valid)
{
  if (!valid) return 0.f;
  // K gathers: default (regular-temporal) policy -- this is the only stream
  // with reuse, so it owns the L2.
  const v4f* kp = (const v4f*)(Kb + (size_t)(yc * W_ + xc) * C_);
  float s = 0.f;
#pragma unroll
  for (int i = 0; i < 8; ++i) {
    v4f k4 = kp[i];
    s += q[i].x * k4.x + q[i].y * k4.y + q[i].z * k4.z + q[i].w * k4.w;
  }
  return wgt * s;
}

__global__ void __launch_bounds__(256) cost_volume_kernel(
    const float* __restrict__ ws,
    const float* __restrict__ d1, const float* __restrict__ d2,
    float* __restrict__ out)
{
  const int pix = blockIdx.x * 256 + threadIdx.x;   // 0 .. NPIX_-1
  const int dir = blockIdx.y;                       // 0: (Qx,Ky,d1) 1: (Qy,Kx,d2)
  const int b   = pix >> 17;
  const int hw  = pix & (HW_ - 1);
  const int h   = hw >> 9;                          // W_ = 2^9

  const float* Qp   = ws + (size_t)(dir ? 2 : 0) * NPC_ + (size_t)pix * C_;
  const float* Kb   = ws + (size_t)(dir ? 3 : 1) * NPC_ + (size_t)b * HW_ * C_;
  const float* disp = (dir ? d2 : d1) + (size_t)b * ND_ * HW_ + hw;
  float* op         = out + (size_t)dir * B_ * ND_ * HW_ + (size_t)b * ND_ * HW_ + hw;

  // Q vector is read exactly once per thread -> non-temporal (last use):
  // evicting it immediately leaves more L2 for the K gathers.
  v4f q[8];
#pragma unroll
  for (int i = 0; i < 8; ++i)
    q[i] = __builtin_nontemporal_load(((const v4f*)Qp) + i);

  // iy = h * H/(H-1) - 0.5  (align_corners=False unnormalization)
  const float iy  = (float)h * ((float)H_ / (float)(H_ - 1)) - 0.5f;
  const float y0f = floorf(iy);
  const float wy1 = iy - y0f, wy0 = 1.f - wy1;
  const int y0 = (int)y0f, y1 = y0 + 1;
  const bool vy0 = (y0 >= 0) && (y0 < H_);
  const bool vy1 = (y1 >= 0) && (y1 < H_);

  for (int d = 0; d < ND_; ++d) {
    const float dx  = __builtin_nontemporal_load(disp + (size_t)d * HW_);
    const float ix  = dx * ((float)W_ / (float)(W_ - 1)) - 0.5f;
    const float x0f = floorf(ix);
    const float wx1 = ix - x0f, wx0 = 1.f - wx1;
    const int x0 = (int)x0f, x1 = x0 + 1;
    const bool vx0 = (x0 >= 0) && (x0 < W_);
    const bool vx1 = (x1 >= 0) && (x1 < W_);

    float acc = 0.f;
    acc += corner_dot(Kb, q, y0, x0, wy0 * wx0, vy0 && vx0);
    acc += corner_dot(Kb, q, y0, x1, wy0 * wx1, vy0 && vx1);
    acc += corner_dot(Kb, q, y1, x0, wy1 * wx0, vy1 && vx0);
    acc += corner_dot(Kb, q, y1, x1, wy1 * wx1, vy1 && vx1);

    // Output is write-once, never re-read -> non-temporal store so the
    // 50 MB of cost-volume dirty lines don't evict K from L2.
    __builtin_nontemporal_store(acc * (1.0f / (float)C_), op + (size_t)d * HW_);
  }
}

// ---------------------------------------------------------------------------
// Entry point. Needs ws_size >= 4 * NPC_ * 4 bytes = 134,217,728 bytes.
// ---------------------------------------------------------------------------
extern "C" void kernel_launch(void* const* d_in, const int* in_sizes, int n_in,
                              void* d_out, int out_size, void* d_ws, size_t ws_size,
                              hipStream_t stream)
{
  const float* x  = (const float*)d_in[0];
  const float* y  = (const float*)d_in[1];
  const float* d1 = (const float*)d_in[2];
  const float* d2 = (const float*)d_in[3];
  const float* Wq = (const float*)d_in[4];
  const float* bq = (const float*)d_in[5];
  const float* Wk = (const float*)d_in[6];
  const float* bk = (const float*)d_in[7];
  float* out = (float*)d_out;
  float* ws  = (float*)d_ws;
  (void)in_sizes; (void)n_in; (void)out_size; (void)ws_size;

  // Stage 1: 2048 blocks x 256 threads; each wave handles 16 pixels,
  // 64 fp32 WMMAs per wave covering all four Q/K tensors.
  conv1x1_wmma_kernel<<<dim3(NPIX_ / 128), dim3(256), 0, stream>>>(
      x, y, Wq, bq, Wk, bk, ws);

  // Stage 2: one thread per (b,h,w), blockIdx.y = direction.
  cost_volume_kernel<<<dim3(NPIX_ / 256, 2), dim3(256), 0, stream>>>(
      ws, d1, d2, out);
}